// ChannelRegMixer_18631568130208
// MI455X (gfx1250) — compile-verified
//
#include <hip/hip_runtime.h>
#include <math.h>

// ---------------- problem constants ----------------
constexpr int   PLANE   = 256 * 256;        // 65536
constexpr int   NPLANES = 8 * 64;           // 512
constexpr float EPSV  = 1e-6f;
constexpr float STEPV = 0.25f;
constexpr float BASEL = 0.03f;
constexpr float KINV2 = 1.0f / (0.03f * 0.03f);   // 1/kappa^2

constexpr int SCR = PLANE;                  // scratch base (float index) in LDS
constexpr int LDS_FLOATS = PLANE + 272;
constexpr size_t LDS_BYTES = (size_t)LDS_FLOATS * sizeof(float);

typedef float v2f __attribute__((ext_vector_type(2)));
typedef float v8f __attribute__((ext_vector_type(8)));

__device__ __forceinline__ float frcp (float x) { return __builtin_amdgcn_rcpf(x);  }
__device__ __forceinline__ float fsqrt(float x) { return __builtin_amdgcn_sqrtf(x); }

// TV normalization n = g/(|g|+eps)  (g==0 -> 0, covers all pad cases)
__device__ __forceinline__ float tvn(float g) { return g * frcp(fabsf(g) + EPSV); }
// AD scalar factor p = g(mag)/ (mag+eps), with mag^2 = gx^2+gy^2+EPS
__device__ __forceinline__ float adp(float gx, float gy) {
  float s = fmaf(gx, gx, fmaf(gy, gy, EPSV));
  return frcp(fmaf(s, KINV2, 1.f)) * frcp(fsqrt(s) + EPSV);
}

// fused TV+Lap+AD update for a 4-pixel quad at linear index q (j aligned to 4).
// Shares the column flux evaluations: AD/TV x-flux at columns j-1..j+3 serves
// both the "center" and "left" terms of all four pixels.
__device__ __forceinline__ float4 quad_step(const float* u, int q,
    float w_tv, float w_lap, float w_ad, float slam)
{
  const int i = q >> 8, j = q & 255, row = i << 8;
  const bool hU = (i > 0), hD = (i < 255), hL = (j > 0), hR4 = (j < 252);

  float4 c4 = *(const float4*)(u + row + j);
  float lft = hL  ? u[row + j - 1] : 0.f;
  float rr  = hR4 ? u[row + j + 4] : 0.f;
  float4 up4 = make_float4(0.f, 0.f, 0.f, 0.f); float ur = 0.f;
  if (hU) { up4 = *(const float4*)(u + row - 256 + j);
            ur  = hR4 ? u[row - 256 + j + 4] : 0.f; }
  float4 dn4 = make_float4(0.f, 0.f, 0.f, 0.f); float dl = 0.f;
  if (hD) { dn4 = *(const float4*)(u + row + 256 + j);
            dl  = hL ? u[row + 256 + j - 1] : 0.f; }

  // ---- row i, columns j-1 .. j+3 (t = 0..4): forward grads at each column ----
  float gx0 = hL  ? (c4.x - lft)  : 0.f;
  float gx1 = c4.y - c4.x;
  float gx2 = c4.z - c4.y;
  float gx3 = c4.w - c4.z;
  float gx4 = hR4 ? (rr - c4.w)   : 0.f;

  float gy0 = (hD && hL) ? (dl - lft) : 0.f;
  float gy1 = hD ? (dn4.x - c4.x) : 0.f;
  float gy2 = hD ? (dn4.y - c4.y) : 0.f;
  float gy3 = hD ? (dn4.z - c4.z) : 0.f;
  float gy4 = hD ? (dn4.w - c4.w) : 0.f;

  // TV x-flux at columns j-1..j+3 ; TV y-flux at columns j..j+3
  float tx0 = tvn(gx0), tx1 = tvn(gx1), tx2 = tvn(gx2), tx3 = tvn(gx3), tx4 = tvn(gx4);
  float ty1 = tvn(gy1), ty2 = tvn(gy2), ty3 = tvn(gy3), ty4 = tvn(gy4);

  // AD fluxes at columns j-1..j+3
  float p0 = adp(gx0, gy0), p1 = adp(gx1, gy1), p2 = adp(gx2, gy2),
        p3 = adp(gx3, gy3), p4 = adp(gx4, gy4);
  float ax0 = p0 * gx0, ax1 = p1 * gx1, ax2 = p2 * gx2, ax3 = p3 * gx3, ax4 = p4 * gx4;
  float ay1 = p1 * gy1, ay2 = p2 * gy2, ay3 = p3 * gy3, ay4 = p4 * gy4;

  // ---- row i-1, columns j..j+3 (m = 0..3): y-fluxes from the row above ----
  float gyU0 = hU ? (c4.x - up4.x) : 0.f;
  float gyU1 = hU ? (c4.y - up4.y) : 0.f;
  float gyU2 = hU ? (c4.z - up4.z) : 0.f;
  float gyU3 = hU ? (c4.w - up4.w) : 0.f;
  float gxU0 = hU ? (up4.y - up4.x) : 0.f;
  float gxU1 = hU ? (up4.z - up4.y) : 0.f;
  float gxU2 = hU ? (up4.w - up4.z) : 0.f;
  float gxU3 = (hU && hR4) ? (ur - up4.w) : 0.f;
  float tyU0 = tvn(gyU0), tyU1 = tvn(gyU1), tyU2 = tvn(gyU2), tyU3 = tvn(gyU3);
  float ayU0 = adp(gxU0, gyU0) * gyU0;
  float ayU1 = adp(gxU1, gyU1) * gyU1;
  float ayU2 = adp(gxU2, gyU2) * gyU2;
  float ayU3 = adp(gxU3, gyU3) * gyU3;

  // ---- 5-point Laplacian with zero padding (OOB values pre-zeroed) ----
  float lap0 = (lft  + c4.y) + (up4.x + dn4.x) - 4.f * c4.x;
  float lap1 = (c4.x + c4.z) + (up4.y + dn4.y) - 4.f * c4.y;
  float lap2 = (c4.y + c4.w) + (up4.z + dn4.z) - 4.f * c4.z;
  float lap3 = (c4.z + rr  ) + (up4.w + dn4.w) - 4.f * c4.w;

  float4 o;
  {
    float dtv = (tx0 - tx1) + (tyU0 - ty1);
    float dad = (ax0 - ax1) + (ayU0 - ay1);
    o.x = c4.x - slam * (w_tv * dtv + w_lap * lap0 + w_ad * dad);
  }
  {
    float dtv = (tx1 - tx2) + (tyU1 - ty2);
    float dad = (ax1 - ax2) + (ayU1 - ay2);
    o.y = c4.y - slam * (w_tv * dtv + w_lap * lap1 + w_ad * dad);
  }
  {
    float dtv = (tx2 - tx3) + (tyU2 - ty3);
    float dad = (ax2 - ax3) + (ayU2 - ay3);
    o.z = c4.z - slam * (w_tv * dtv + w_lap * lap2 + w_ad * dad);
  }
  {
    float dtv = (tx3 - tx4) + (tyU3 - ty4);
    float dad = (ax3 - ax4) + (ayU3 - ay4);
    o.w = c4.w - slam * (w_tv * dtv + w_lap * lap3 + w_ad * dad);
  }
  return o;
}

extern "C" __global__ void __launch_bounds__(1024, 1)
crm_fused(const float* __restrict__ x,
          const float* __restrict__ mlp_w,
          const float* __restrict__ mlp_b,
          float* __restrict__ out,
          float* __restrict__ out_mix,
          float* __restrict__ out_lam)
{
  extern __shared__ float u[];                     // [PLANE] plane + [272] scratch
  const int tid   = threadIdx.x;
  const int lane  = tid & 31;
  const int wv    = tid >> 5;
  const int plane = blockIdx.x;
  const long base = (long)plane * PLANE;
  float* scr = u + SCR;

  // ---- 1) async DMA: global plane -> LDS (b128 per lane, fully coalesced) ----
  {
    const float* g = x + base;
    unsigned int l0 = (unsigned int)(unsigned long long)(size_t)u; // low32 = LDS offset
    #pragma unroll
    for (int k = 0; k < 16; ++k) {
      int q = (tid << 2) + (k << 12);
      unsigned long long ga = (unsigned long long)(size_t)(g + q);
      unsigned int la = l0 + (unsigned int)(q << 2);
      asm volatile("global_load_async_to_lds_b128 %0, %1, off"
                   :: "v"(la), "v"(ga) : "memory");
    }
    asm volatile("s_wait_asynccnt 0x0" ::: "memory");
  }
  __syncthreads();

  // ---- 2) per-plane stats from LDS ----
  float sum = 0.f, mx = -3.402823466e38f, sgx = 0.f, sgy = 0.f, slap = 0.f;
  {
    auto stat_px = [&](float c, float L, float R, float U, float D, bool hR, bool hD) {
      sum += c; mx = fmaxf(mx, c);
      float gxr = hR ? (R - c) : 0.f;
      float gyd = hD ? (D - c) : 0.f;
      sgx += fabsf(gxr); sgy += fabsf(gyd);
      slap += fabsf((L + R) + (U + D) - 4.f * c);
    };
    #pragma unroll 1
    for (int k = 0; k < 16; ++k) {
      const int q = (tid << 2) + (k << 12);
      const int i = q >> 8, j = q & 255, row = i << 8;
      float4 c4 = *(const float4*)(u + row + j);
      float lft = (j > 0)   ? u[row + j - 1] : 0.f;
      float rr  = (j < 252) ? u[row + j + 4] : 0.f;
      float4 up4 = make_float4(0.f, 0.f, 0.f, 0.f);
      if (i > 0)   up4 = *(const float4*)(u + row - 256 + j);
      float4 dn4 = make_float4(0.f, 0.f, 0.f, 0.f);
      if (i < 255) dn4 = *(const float4*)(u + row + 256 + j);
      const bool hD = (i < 255);
      stat_px(c4.x, lft , c4.y, up4.x, dn4.x, true   , hD);
      stat_px(c4.y, c4.x, c4.z, up4.y, dn4.y, true   , hD);
      stat_px(c4.z, c4.y, c4.w, up4.z, dn4.z, true   , hD);
      stat_px(c4.w, c4.z, rr  , up4.w, dn4.w, j < 252, hD);
    }
  }
  #pragma unroll
  for (int off = 16; off > 0; off >>= 1) {       // wave32 tree reduce
    sum  += __shfl_down(sum,  off, 32);
    sgx  += __shfl_down(sgx,  off, 32);
    sgy  += __shfl_down(sgy,  off, 32);
    slap += __shfl_down(slap, off, 32);
    mx    = fmaxf(mx, __shfl_down(mx, off, 32));
  }
  if (lane == 0) {
    scr[wv * 8 + 0] = sum;  scr[wv * 8 + 1] = mx;
    scr[wv * 8 + 2] = sgx;  scr[wv * 8 + 3] = sgy; scr[wv * 8 + 4] = slap;
  }
  __syncthreads();
  if (tid == 0) {
    float s0 = 0.f, s1 = -3.402823466e38f, s2 = 0.f, s3 = 0.f, s4 = 0.f;
    for (int w2 = 0; w2 < 32; ++w2) {
      s0 += scr[w2 * 8 + 0]; s1 = fmaxf(s1, scr[w2 * 8 + 1]);
      s2 += scr[w2 * 8 + 2]; s3 += scr[w2 * 8 + 3]; s4 += scr[w2 * 8 + 4];
    }
    scr[256 + 0] = s0 * (1.f / 65536.f);                                  // gap
    scr[256 + 1] = s1;                                                    // gmp
    scr[256 + 2] = 0.5f * (s2 * (1.f / (256.f * 255.f)) +
                           s3 * (1.f / (255.f * 256.f)));                 // gmean
    scr[256 + 3] = s4 * (1.f / 65536.f);                                  // lmean
  }
  __syncthreads();

  // ---- 3) 4x4 MLP head via V_WMMA_F32_16X16X4_F32 (wave 0, EXEC all-ones) ----
  if (tid < 32) {
    // A (16x4 f32): VGPR0 = K0 (lanes 0-15) / K2 (lanes 16-31); VGPR1 = K1 / K3.
    float a0 = 0.f, a1 = 0.f;
    if (lane == 0)       { a0 = scr[256 + 0]; a1 = scr[256 + 1]; }
    else if (lane == 16) { a0 = scr[256 + 2]; a1 = scr[256 + 3]; }
    // B (4x16 f32): B[k][n] = mlp_w[n*4+k]; same half-lane K striping as A.
    float b0 = 0.f, b1 = 0.f;
    int col = lane & 15;
    if (col < 4) {
      int krow = (lane >> 4) << 1;           // 0 or 2
      b0 = mlp_w[col * 4 + krow];
      b1 = mlp_w[col * 4 + krow + 1];
    }
    v2f A; A.x = a0; A.y = a1;
    v2f B; B.x = b0; B.y = b1;
    v8f C = {0.f, 0.f, 0.f, 0.f, 0.f, 0.f, 0.f, 0.f};
    v8f D = __builtin_amdgcn_wmma_f32_16x16x4_f32(false, A, false, B,
                                                  (short)0, C, false, false);
    if (lane < 4) scr[260 + lane] = D[0] + mlp_b[lane];   // D[0]: M=0, N=lane
  }
  __syncthreads();

  // ---- 4) softmax / softplus -> broadcast mix weights + lambda ----
  if (tid == 0) {
    float h0 = scr[260], h1 = scr[261], h2 = scr[262], h3 = scr[263];
    float m  = fmaxf(h0, fmaxf(h1, h2));
    float e0 = __expf(h0 - m), e1 = __expf(h1 - m), e2 = __expf(h2 - m);
    float inv = frcp(e0 + e1 + e2);
    float w0 = e0 * inv, w1 = e1 * inv, w2 = e2 * inv;
    float sp  = fmaxf(h3, 0.f) + log1pf(__expf(-fabsf(h3)));
    float lam = BASEL * sp;
    scr[264] = w0; scr[265] = w1; scr[266] = w2; scr[267] = STEPV * lam;
    out_mix[plane * 3 + 0] = w0;
    out_mix[plane * 3 + 1] = w1;
    out_mix[plane * 3 + 2] = w2;
    out_lam[plane] = lam;
  }
  __syncthreads();
  const float w_tv = scr[264], w_lap = scr[265], w_ad = scr[266], slam = scr[267];

  // ---- 5) diffusion step 1: banded in-place sweep with a 1-deep register
  //         pipeline. Band b = rows [16b,16b+16) = one quad per thread.
  //         Compute band b (reads original bands b-1..b+1), barrier (all reads
  //         of band b-1 done), then flush band b-1's result into the plane. ----
  {
    float4 prev = make_float4(0.f, 0.f, 0.f, 0.f);
    #pragma unroll 1
    for (int b = 0; b < 16; ++b) {
      const int q = (tid << 2) + (b << 12);
      float4 r = quad_step(u, q, w_tv, w_lap, w_ad, slam);
      __syncthreads();
      if (b > 0) *(float4*)(u + q - 4096) = prev;
      prev = r;
    }
    __syncthreads();
    *(float4*)(u + (tid << 2) + (15 << 12)) = prev;   // flush last band
  }
  __syncthreads();

  // ---- 6) diffusion step 2: LDS -> global out (coalesced b128 stores) ----
  #pragma unroll 1
  for (int b = 0; b < 16; ++b) {
    const int q = (tid << 2) + (b << 12);
    float4 o = quad_step(u, q, w_tv, w_lap, w_ad, slam);
    *(float4*)(out + base + q) = o;
  }
}

extern "C" void kernel_launch(void* const* d_in, const int* in_sizes, int n_in,
                              void* d_out, int out_size, void* d_ws, size_t ws_size,
                              hipStream_t stream) {
  (void)in_sizes; (void)n_in; (void)out_size; (void)d_ws; (void)ws_size;
  const float* x  = (const float*)d_in[0];
  const float* mw = (const float*)d_in[1];
  const float* mb = (const float*)d_in[2];
  float* out      = (float*)d_out;
  float* out_mix  = out + (size_t)NPLANES * PLANE;   // [8,64,3]
  float* out_lam  = out_mix + NPLANES * 3;           // [8,64,1,1]
  hipFuncSetAttribute(reinterpret_cast<const void*>(crm_fused),
                      hipFuncAttributeMaxDynamicSharedMemorySize, (int)LDS_BYTES);
  crm_fused<<<NPLANES, 1024, LDS_BYTES, stream>>>(x, mw, mb, out, out_mix, out_lam);
}